// FastTSAGEConv_35227321762436
// MI455X (gfx1250) — compile-verified
//
#include <hip/hip_runtime.h>

typedef __attribute__((ext_vector_type(2))) float v2f;
typedef __attribute__((ext_vector_type(8))) float v8f;

#define F 128
#define BROWS 1024

// ---------------------------------------------------------------------------
// Kernel 1: segment starts. dst_ids is sorted ascending; the first edge of
// each run writes its index into seg[dst]. Entries for absent dst ids are
// never read.
// ---------------------------------------------------------------------------
__global__ __launch_bounds__(256) void seg_start_kernel(
    const int* __restrict__ dst_ids, int* __restrict__ seg, int E) {
  int e = blockIdx.x * 256 + threadIdx.x;
  if (e >= E) return;
  if (e == 0 || dst_ids[e] != dst_ids[e - 1]) seg[dst_ids[e]] = e;
}

// ---------------------------------------------------------------------------
// Kernels 2-4: global column-wise inclusive cumsum of src_feat -> cs.
// One thread per feature column; rows walked serially (coalesced 512B/row).
// ---------------------------------------------------------------------------
__global__ __launch_bounds__(F) void block_partial_kernel(
    const float* __restrict__ src, float* __restrict__ bsum, int E) {
  int f = threadIdx.x;
  long e0 = (long)blockIdx.x * BROWS;
  long e1 = e0 + BROWS; if (e1 > E) e1 = E;
  float s = 0.f;
  for (long e = e0; e < e1; ++e) s += src[e * F + f];
  bsum[(long)blockIdx.x * F + f] = s;
}

__global__ __launch_bounds__(F) void block_scan_kernel(
    float* __restrict__ bsum, int NB) {
  int f = threadIdx.x;
  float run = 0.f;
  for (int b = 0; b < NB; ++b) {
    float t = bsum[(long)b * F + f];
    bsum[(long)b * F + f] = run;   // exclusive prefix
    run += t;
  }
}

__global__ __launch_bounds__(F) void cumsum_kernel(
    const float* __restrict__ src, const float* __restrict__ bsum,
    float* __restrict__ cs, int E) {
  int f = threadIdx.x;
  long e0 = (long)blockIdx.x * BROWS;
  long e1 = e0 + BROWS; if (e1 > E) e1 = E;
  float run = bsum[(long)blockIdx.x * F + f];
  for (long e = e0; e < e1; ++e) {
    run += src[e * F + f];
    cs[e * F + f] = run;
  }
}

// ---------------------------------------------------------------------------
// Kernel 5: fused gather + dual GEMM via V_WMMA_F32_16X16X4_F32.
// Block = 256 threads = 8 wave32, handles 64 edges x 128 out cols.
// Wave w owns output columns [16w, 16w+16); loops 4 M-tiles of 16 edges.
// Two independent accumulator chains (self / neigh) so the matrix pipe always
// has a WMMA whose C-input is not the immediately preceding D.
// ---------------------------------------------------------------------------
__global__ __launch_bounds__(256) void sage_gemm_kernel(
    const float* __restrict__ dst_feat, const float* __restrict__ cs,
    const int* __restrict__ dst_ids, const int* __restrict__ dst_max_eid,
    const float* __restrict__ dst_deg, const int* __restrict__ seg,
    const float* __restrict__ W_self, const float* __restrict__ b_self,
    const float* __restrict__ W_neigh, const float* __restrict__ b_neigh,
    float* __restrict__ out, int E) {
  __shared__ float As[64][F + 4];  // dst_feat tile (pad -> no bank conflicts)
  __shared__ float An[64][F + 4];  // neighbor-mean tile

  const int tid  = threadIdx.x;
  const int wave = tid >> 5;
  const int lane = tid & 31;
  const long e0  = (long)blockIdx.x * 64;

  // --- stage A_self tile ---
  for (int i = tid; i < 64 * (F / 4); i += 256) {
    int r  = i >> 5;            // row 0..63
    int c4 = (i & 31) * 4;      // col 0,4,...,124
    float4 v = make_float4(0.f, 0.f, 0.f, 0.f);
    long e = e0 + r;
    if (e < E) v = *reinterpret_cast<const float4*>(dst_feat + e * F + c4);
    *reinterpret_cast<float4*>(&As[r][c4]) = v;
  }
  // --- stage A_neigh tile: (cs[m] - cs[s-1]) / (deg + 1) ---
  for (int i = tid; i < 64 * (F / 4); i += 256) {
    int r  = i >> 5;
    int c4 = (i & 31) * 4;
    float4 v = make_float4(0.f, 0.f, 0.f, 0.f);
    long e = e0 + r;
    if (e < E) {
      int   m   = dst_max_eid[e];
      int   s   = seg[dst_ids[e]];
      float inv = 1.0f / (dst_deg[e] + 1.0f);
      float4 a = *reinterpret_cast<const float4*>(cs + (long)m * F + c4);
      float4 p = make_float4(0.f, 0.f, 0.f, 0.f);
      if (s > 0) p = *reinterpret_cast<const float4*>(cs + (long)(s - 1) * F + c4);
      v.x = (a.x - p.x) * inv;
      v.y = (a.y - p.y) * inv;
      v.z = (a.z - p.z) * inv;
      v.w = (a.w - p.w) * inv;
    }
    *reinterpret_cast<float4*>(&An[r][c4]) = v;
  }
  __syncthreads();

  const int n0    = wave * 16;
  const int nlane = lane & 15;
  const int khalf = (lane < 16) ? 0 : 2;  // ISA A/B frag layout: upper half-wave holds K+2

  // B fragment sets; compiler may keep resident or refetch from L2 per M-tile.
  v2f Bs[32], Bn[32];
#pragma unroll
  for (int s = 0; s < 32; ++s) {
    int k = 4 * s + khalf;
    v2f b0, b1;
    b0.x = W_self [(long)k * F + n0 + nlane];
    b0.y = W_self [(long)(k + 1) * F + n0 + nlane];
    b1.x = W_neigh[(long)k * F + n0 + nlane];
    b1.y = W_neigh[(long)(k + 1) * F + n0 + nlane];
    Bs[s] = b0;
    Bn[s] = b1;
  }
  const float bias = b_self[n0 + nlane] + b_neigh[n0 + nlane];

  for (int mt = 0; mt < 4; ++mt) {
    const int r = mt * 16 + nlane;  // A row for this lane (lanes 0-15 / 16-31 share M)
    v8f acc_s, acc_n;
#pragma unroll
    for (int v = 0; v < 8; ++v) { acc_s[v] = bias; acc_n[v] = 0.f; }

#pragma unroll
    for (int s = 0; s < 32; ++s) {
      int k = 4 * s + khalf;
      v2f a;
      a.x = As[r][k];
      a.y = As[r][k + 1];
      acc_s = __builtin_amdgcn_wmma_f32_16x16x4_f32(
          false, a, false, Bs[s], (short)0, acc_s, false, false);
      v2f an;
      an.x = An[r][k];
      an.y = An[r][k + 1];
      acc_n = __builtin_amdgcn_wmma_f32_16x16x4_f32(
          false, an, false, Bn[s], (short)0, acc_n, false, false);
    }

    // D layout: VGPR v -> M = v (+8 for lanes 16-31), N = lane & 15.
    const int mbase = mt * 16 + ((lane < 16) ? 0 : 8);
#pragma unroll
    for (int v = 0; v < 8; ++v) {
      long e = e0 + mbase + v;
      if (e < E) out[e * F + n0 + nlane] = acc_s[v] + acc_n[v];
    }
  }
}

// ---------------------------------------------------------------------------
// Host launch. Inputs (setup_inputs order):
//  0 src_feat [E*128] f32   1 dst_feat [E*128] f32   2 dst_ids [E] i32
//  3 dst_max_eid [E] i32    4 dst_deg [E] f32
//  5 W_self [128*128] f32   6 b_self [128] f32
//  7 W_neigh [128*128] f32  8 b_neigh [128] f32
// Workspace: cs[E*128] f32 | bsum[NB*128] f32 | seg[E] i32
// ---------------------------------------------------------------------------
extern "C" void kernel_launch(void* const* d_in, const int* in_sizes, int n_in,
                              void* d_out, int out_size, void* d_ws, size_t ws_size,
                              hipStream_t stream) {
  const float* src_feat    = (const float*)d_in[0];
  const float* dst_feat    = (const float*)d_in[1];
  const int*   dst_ids     = (const int*)d_in[2];
  const int*   dst_max_eid = (const int*)d_in[3];
  const float* dst_deg     = (const float*)d_in[4];
  const float* W_self      = (const float*)d_in[5];
  const float* b_self      = (const float*)d_in[6];
  const float* W_neigh     = (const float*)d_in[7];
  const float* b_neigh     = (const float*)d_in[8];

  const int E  = in_sizes[2];
  const int NB = (E + BROWS - 1) / BROWS;

  char*  ws   = (char*)d_ws;
  float* cs   = (float*)ws;
  size_t off  = (size_t)E * F * sizeof(float);
  float* bsum = (float*)(ws + off);
  off += (size_t)NB * F * sizeof(float);
  int*   seg  = (int*)(ws + off);

  float* out = (float*)d_out;

  seg_start_kernel<<<(E + 255) / 256, 256, 0, stream>>>(dst_ids, seg, E);
  block_partial_kernel<<<NB, F, 0, stream>>>(src_feat, bsum, E);
  block_scan_kernel<<<1, F, 0, stream>>>(bsum, NB);
  cumsum_kernel<<<NB, F, 0, stream>>>(src_feat, bsum, cs, E);
  sage_gemm_kernel<<<(E + 63) / 64, 256, 0, stream>>>(
      dst_feat, cs, dst_ids, dst_max_eid, dst_deg, seg,
      W_self, b_self, W_neigh, b_neigh, out, E);
}